// RelationProbe_76897094467881
// MI455X (gfx1250) — compile-verified
//
#include <hip/hip_runtime.h>

// ---------------------------------------------------------------------------
// RelationProbe on MI455X (gfx1250, wave32).
// out[r, row] = dot(z[row, :], W[r, pair_idx[row], :]) + b[r, pair_idx[row]]
//
// Strategy: dense 24-head GEMM per 16-row tile via V_WMMA_F32_16X16X4_F32
// (f32 in / f32 accumulate -> numerics match the f32 einsum), then per-row
// head selection from LDS. Bandwidth-bound: reads z exactly once, coalesced.
// ---------------------------------------------------------------------------

typedef __attribute__((ext_vector_type(2))) float v2f;
typedef __attribute__((ext_vector_type(4))) float v4f;
typedef __attribute__((ext_vector_type(8))) float v8f;

#define DIM     64
#define NREL    4
#define NPAIR   6
#define NHEAD   24            // NREL * NPAIR, padded to 32 columns for WMMA
#define ZSTRIDE 68            // LDS stride (floats) for z tile: 16B aligned, bank-friendly
#define LSTRIDE 36            // LDS stride (floats) for logits tile: conflict-free C/D spill
#define WAVES   8             // waves per 256-thread block

__global__ __launch_bounds__(256)
void relprobe_wmma_f32(const float* __restrict__ z,
                       const float* __restrict__ W,
                       const float* __restrict__ bias,
                       const int*   __restrict__ pair_idx,
                       float*       __restrict__ out,
                       int nrows, int ntiles)
{
    __shared__ float smem[WAVES * (16 * ZSTRIDE + 16 * LSTRIDE)];

    const int lane = threadIdx.x & 31;
    const int wave = threadIdx.x >> 5;
    float* zs = smem + wave * (16 * ZSTRIDE + 16 * LSTRIDE);
    float* ls = zs + 16 * ZSTRIDE;

    const int nlo = lane & 15;   // matrix row (A) / column (B,C,D) within tile
    const int hi  = lane >> 4;   // K-half selector for A/B fragments

    // ---- B fragments: W padded to a 64x32 column block, loaded once per wave.
    // Lane holds B[k = 4c + 2*hi + v, n = t*16 + nlo], heads n>=24 are zero.
    v2f bfrag[2][16];
#pragma unroll
    for (int t = 0; t < 2; ++t) {
        const int n = t * 16 + nlo;
#pragma unroll
        for (int c = 0; c < 16; ++c) {
            if (n < NHEAD) {
                bfrag[t][c] = *(const v2f*)(W + n * DIM + 4 * c + 2 * hi);
            } else {
                bfrag[t][c] = (v2f){0.0f, 0.0f};
            }
        }
    }

    // ---- Grid-stride over 16-row tiles (tile index is wave-uniform -> EXEC
    // stays all-ones for WMMA; each wave owns a private LDS region, so LDS
    // in-order-within-wave semantics make barriers unnecessary).
    const int waveId   = blockIdx.x * WAVES + wave;
    const int waveStep = gridDim.x * WAVES;

    for (int tile = waveId; tile < ntiles; tile += waveStep) {
        const long long rowBase = (long long)tile * 16;

        // Stage the 16x64 f32 z tile into LDS, fully coalesced b128 loads.
        const float* ztile = z + rowBase * DIM;
#pragma unroll
        for (int i = 0; i < 8; ++i) {
            const int flat = i * 32 + lane;      // 256 float4 per tile
            const int row  = flat >> 4;
            const int c4   = flat & 15;
            v4f v = *(const v4f*)(ztile + flat * 4);
            *(v4f*)(zs + row * ZSTRIDE + c4 * 4) = v;
        }

        // 16 K-chunks x 2 N-tiles of V_WMMA_F32_16X16X4_F32.
        v8f acc0 = {};
        v8f acc1 = {};
#pragma unroll
        for (int c = 0; c < 16; ++c) {
            v2f a = *(const v2f*)(zs + nlo * ZSTRIDE + 4 * c + 2 * hi);
            acc0 = __builtin_amdgcn_wmma_f32_16x16x4_f32(
                false, a, false, bfrag[0][c], (short)0, acc0, false, false);
            acc1 = __builtin_amdgcn_wmma_f32_16x16x4_f32(
                false, a, false, bfrag[1][c], (short)0, acc1, false, false);
        }

        // Spill logits tile (16 rows x 24 heads, stride-padded) to LDS.
        // C/D layout: VGPR g, lane L -> M = g + 8*(L>=16), N = L%16.
#pragma unroll
        for (int g = 0; g < 8; ++g) {
            const int m = g + 8 * hi;
            ls[m * LSTRIDE + nlo]      = acc0[g];
            ls[m * LSTRIDE + 16 + nlo] = acc1[g];
        }

        // Select 4 heads per row via pair_idx, add bias, coalesced stores.
        // 64 outputs per tile -> 2 per lane: j = r*16 + m.
#pragma unroll
        for (int half = 0; half < 2; ++half) {
            const int j = half * 32 + lane;
            const int r = j >> 4;
            const int m = j & 15;
            const long long row = rowBase + m;
            const int p = pair_idx[row];
            const int h = r * NPAIR + p;
            out[(long long)r * nrows + row] = ls[m * LSTRIDE + h] + bias[h];
        }
    }
}

extern "C" void kernel_launch(void* const* d_in, const int* in_sizes, int n_in,
                              void* d_out, int out_size, void* d_ws, size_t ws_size,
                              hipStream_t stream)
{
    const float* z    = (const float*)d_in[0];   // (B, 64) f32
    const float* W    = (const float*)d_in[1];   // (4, 6, 64) f32
    const float* bias = (const float*)d_in[2];   // (4, 6) f32
    const int*   pidx = (const int*)d_in[3];     // (B,) int
    float*       out  = (float*)d_out;           // (4, B) f32

    const int nrows  = in_sizes[0] / DIM;        // B (multiple of 16 here)
    const int ntiles = nrows / 16;

    int blocks = (ntiles + WAVES - 1) / WAVES;
    if (blocks > 4096) blocks = 4096;            // grid-stride; amortize W loads
    if (blocks < 1) blocks = 1;

    relprobe_wmma_f32<<<blocks, 256, 0, stream>>>(z, W, bias, pidx, out,
                                                  nrows, ntiles);
}